// SingleChannelConv_21474836480482
// MI455X (gfx1250) — compile-verified
//
#include <hip/hip_runtime.h>
#include <hip/hip_bf16.h>

typedef __attribute__((ext_vector_type(2))) float v2f;
typedef __attribute__((ext_vector_type(8))) float v8f;

#define C_IN   96
#define C_OUT  128
#define KOFF   27
#define BN_EPS 1e-5f

// ---------------------------------------------------------------------------
// Kernel 1: X1 = feats @ w1 ; R = feats @ w_down  (dual accumulators, shared
// A-tile in LDS). Also accumulates per-channel sum / sumsq of X1 for BN1.
// grid = N/16 blocks, 256 threads (8 waves); wave w -> output cols [16w,16w+16)
// ---------------------------------------------------------------------------
__global__ void __launch_bounds__(256)
gemm_dual_k(const float* __restrict__ feats,
            const float* __restrict__ w1,
            const float* __restrict__ wdown,
            float* __restrict__ X1, float* __restrict__ R,
            float* __restrict__ sum1, float* __restrict__ sumsq1, int Ntot) {
    __shared__ float As[16][100];              // 16 x 96 A tile, padded stride
    const int t    = threadIdx.x;
    const int row0 = blockIdx.x * 16;

    for (int idx = t; idx < 16 * C_IN; idx += 256) {
        int r = idx / C_IN, c = idx % C_IN;
        int row = row0 + r; if (row >= Ntot) row = Ntot - 1;
        As[r][c] = feats[(size_t)row * C_IN + c];
    }
    __syncthreads();

    const int wave = t >> 5, lane = t & 31, half = lane >> 4, wi = lane & 15;
    const int col0 = wave * 16;

    v8f acc0 = {};   // w1 path
    v8f acc1 = {};   // w_down path
    for (int kk = 0; kk < C_IN / 4; ++kk) {
        const int k0 = 4 * kk + 2 * half;
        v2f a;  a[0]  = As[wi][k0];                    a[1]  = As[wi][k0 + 1];
        v2f b0; b0[0] = w1[k0 * C_OUT + col0 + wi];    b0[1] = w1[(k0 + 1) * C_OUT + col0 + wi];
        v2f b1; b1[0] = wdown[k0 * C_OUT + col0 + wi]; b1[1] = wdown[(k0 + 1) * C_OUT + col0 + wi];
        acc0 = __builtin_amdgcn_wmma_f32_16x16x4_f32(false, a, false, b0, (short)0, acc0, false, false);
        acc1 = __builtin_amdgcn_wmma_f32_16x16x4_f32(false, a, false, b1, (short)0, acc1, false, false);
    }

    float s = 0.f, q = 0.f;
    const int c = col0 + wi;
#pragma unroll
    for (int j = 0; j < 8; ++j) {
        int row = row0 + j + 8 * half;
        if (row < Ntot) {
            float v = acc0[j];
            X1[(size_t)row * C_OUT + c] = v;
            R [(size_t)row * C_OUT + c] = acc1[j];
            s += v; q += v * v;
        }
    }
    atomicAdd(&sum1[c], s);
    atomicAdd(&sumsq1[c], q);
}

// ---------------------------------------------------------------------------
// Fold sum/sumsq into BN scale/bias:  y = x*scale + bias
// ---------------------------------------------------------------------------
__global__ void bn_params_k(const float* __restrict__ sum, const float* __restrict__ sumsq,
                            const float* __restrict__ gamma, const float* __restrict__ beta,
                            float* __restrict__ scale, float* __restrict__ bias, float invN) {
    int c = threadIdx.x;
    float mean = sum[c] * invN;
    float var  = sumsq[c] * invN - mean * mean;
    float s    = gamma[c] * rsqrtf(var + BN_EPS);
    scale[c] = s;
    bias[c]  = beta[c] - mean * s;
}

// ---------------------------------------------------------------------------
// Kernel 3: H2 = (relu(X1*scale1+bias1)) @ w2.  BN+ReLU applied while staging
// the A tile into LDS; w2 (64 KB) streams from L2.
// ---------------------------------------------------------------------------
__global__ void __launch_bounds__(256)
bn_relu_gemm2_k(const float* __restrict__ X1,
                const float* __restrict__ scale1, const float* __restrict__ bias1,
                const float* __restrict__ w2, float* __restrict__ H2, int Ntot) {
    __shared__ float Hs[16][132];
    const int t    = threadIdx.x;
    const int row0 = blockIdx.x * 16;

    for (int idx = t; idx < 16 * C_OUT; idx += 256) {
        int r = idx >> 7, c = idx & 127;
        int row = row0 + r; if (row >= Ntot) row = Ntot - 1;
        float v = X1[(size_t)row * C_OUT + c] * scale1[c] + bias1[c];
        Hs[r][c] = v > 0.f ? v : 0.f;
    }
    __syncthreads();

    const int wave = t >> 5, lane = t & 31, half = lane >> 4, wi = lane & 15;
    const int col0 = wave * 16;

    v8f acc = {};
    for (int kk = 0; kk < C_OUT / 4; ++kk) {
        const int k0 = 4 * kk + 2 * half;
        v2f a; a[0] = Hs[wi][k0];                 a[1] = Hs[wi][k0 + 1];
        v2f b; b[0] = w2[k0 * C_OUT + col0 + wi]; b[1] = w2[(k0 + 1) * C_OUT + col0 + wi];
        acc = __builtin_amdgcn_wmma_f32_16x16x4_f32(false, a, false, b, (short)0, acc, false, false);
    }
#pragma unroll
    for (int j = 0; j < 8; ++j) {
        int row = row0 + j + 8 * half;
        if (row < Ntot) H2[(size_t)row * C_OUT + col0 + wi] = acc[j];
    }
}

// ---------------------------------------------------------------------------
// Kernel 4: channelwise 3x3x3 masked gather conv.
// acc[i,c] = sum_k mask[i,k] * w_cw[k,c] * H2[idx[i,k], c]
// thread = (row, 4-channel group); 8 rows per 256-thread block.
// Per-channel sum/sumsq reduced through LDS, then one global atomic per chan.
// ---------------------------------------------------------------------------
__global__ void __launch_bounds__(256)
gather_cw_k(const float* __restrict__ H2,
            const int* __restrict__ nidx, const unsigned char* __restrict__ nmask,
            const float* __restrict__ wcw, float* __restrict__ ACC,
            float* __restrict__ sum2, float* __restrict__ sumsq2, int Ntot) {
    __shared__ float lsum[C_OUT];
    __shared__ float lsq[C_OUT];
    const int t = threadIdx.x;
    if (t < C_OUT) { lsum[t] = 0.f; lsq[t] = 0.f; }
    __syncthreads();

    const int i  = blockIdx.x * 8 + (t >> 5);
    const int cg = (t & 31) * 4;
    float4 acc = make_float4(0.f, 0.f, 0.f, 0.f);

    if (i < Ntot) {
        const int* ip = nidx + (size_t)i * KOFF;
        const unsigned char* mp = nmask + (size_t)i * KOFF;
#pragma unroll
        for (int k = 0; k < KOFF; ++k) {
            if (mp[k]) {
                const size_t j = (size_t)ip[k];
                const float4 hv = *(const float4*)(H2  + j * C_OUT + cg);
                const float4 wv = *(const float4*)(wcw + k * C_OUT + cg);
                acc.x += hv.x * wv.x; acc.y += hv.y * wv.y;
                acc.z += hv.z * wv.z; acc.w += hv.w * wv.w;
            }
        }
        *(float4*)(ACC + (size_t)i * C_OUT + cg) = acc;
    }
    atomicAdd(&lsum[cg + 0], acc.x); atomicAdd(&lsq[cg + 0], acc.x * acc.x);
    atomicAdd(&lsum[cg + 1], acc.y); atomicAdd(&lsq[cg + 1], acc.y * acc.y);
    atomicAdd(&lsum[cg + 2], acc.z); atomicAdd(&lsq[cg + 2], acc.z * acc.z);
    atomicAdd(&lsum[cg + 3], acc.w); atomicAdd(&lsq[cg + 3], acc.w * acc.w);
    __syncthreads();
    if (t < C_OUT) {
        atomicAdd(&sum2[t], lsum[t]);
        atomicAdd(&sumsq2[t], lsq[t]);
    }
}

// ---------------------------------------------------------------------------
// Kernel 6: out = relu( relu(ACC*scale2+bias2) + R )
// ---------------------------------------------------------------------------
__global__ void __launch_bounds__(256)
final_out_k(const float* __restrict__ ACC, const float* __restrict__ R,
            const float* __restrict__ scale2, const float* __restrict__ bias2,
            float* __restrict__ out, int total4) {
    int idx = blockIdx.x * blockDim.x + threadIdx.x;
    for (; idx < total4; idx += gridDim.x * blockDim.x) {
        const int cg = (idx & 31) * 4;              // 128 ch = 32 float4 per row
        const float4 a = ((const float4*)ACC)[idx];
        const float4 r = ((const float4*)R)[idx];
        float4 o;
        o.x = a.x * scale2[cg + 0] + bias2[cg + 0];
        o.y = a.y * scale2[cg + 1] + bias2[cg + 1];
        o.z = a.z * scale2[cg + 2] + bias2[cg + 2];
        o.w = a.w * scale2[cg + 3] + bias2[cg + 3];
        o.x = (o.x > 0.f ? o.x : 0.f) + r.x;  o.x = o.x > 0.f ? o.x : 0.f;
        o.y = (o.y > 0.f ? o.y : 0.f) + r.y;  o.y = o.y > 0.f ? o.y : 0.f;
        o.z = (o.z > 0.f ? o.z : 0.f) + r.z;  o.z = o.z > 0.f ? o.z : 0.f;
        o.w = (o.w > 0.f ? o.w : 0.f) + r.w;  o.w = o.w > 0.f ? o.w : 0.f;
        ((float4*)out)[idx] = o;
    }
}

extern "C" void kernel_launch(void* const* d_in, const int* in_sizes, int n_in,
                              void* d_out, int out_size, void* d_ws, size_t ws_size,
                              hipStream_t stream) {
    const float* feats  = (const float*)d_in[0];
    const float* w1     = (const float*)d_in[1];
    const float* gamma1 = (const float*)d_in[2];
    const float* beta1  = (const float*)d_in[3];
    const float* w2     = (const float*)d_in[4];
    const float* wcw    = (const float*)d_in[5];
    const float* gamma2 = (const float*)d_in[6];
    const float* beta2  = (const float*)d_in[7];
    const float* wdown  = (const float*)d_in[8];
    const int* nidx     = (const int*)d_in[9];
    const unsigned char* nmask = (const unsigned char*)d_in[10];  // jnp bool = 1 byte
    float* out = (float*)d_out;

    const int Ntot = in_sizes[0] / C_IN;
    const size_t NC = (size_t)Ntot * C_OUT;

    // workspace layout: [X1/ACC | R | H2 | 1024 floats of BN stats]
    float* X1 = (float*)d_ws;
    float* R  = X1 + NC;
    float* H2 = R + NC;
    float* st = H2 + NC;
    float *sum1 = st,        *sumsq1 = st + 128, *scale1 = st + 256, *bias1 = st + 384;
    float *sum2 = st + 512,  *sumsq2 = st + 640, *scale2 = st + 768, *bias2 = st + 896;

    hipMemsetAsync(st, 0, 1024 * sizeof(float), stream);

    const int nb = (Ntot + 15) / 16;
    const float invN = 1.0f / (float)Ntot;

    gemm_dual_k<<<nb, 256, 0, stream>>>(feats, w1, wdown, X1, R, sum1, sumsq1, Ntot);
    bn_params_k<<<1, 128, 0, stream>>>(sum1, sumsq1, gamma1, beta1, scale1, bias1, invN);
    bn_relu_gemm2_k<<<nb, 256, 0, stream>>>(X1, scale1, bias1, w2, H2, Ntot);
    float* ACC = X1;  // reuse: X1 fully consumed by previous kernel
    gather_cw_k<<<(Ntot + 7) / 8, 256, 0, stream>>>(H2, nidx, nmask, wcw, ACC, sum2, sumsq2, Ntot);
    bn_params_k<<<1, 128, 0, stream>>>(sum2, sumsq2, gamma2, beta2, scale2, bias2, invN);
    const int total4 = (int)(NC / 4);
    final_out_k<<<(total4 + 255) / 256, 256, 0, stream>>>(ACC, R, scale2, bias2, out, total4);
}